// LossGenerator_481036337657
// MI455X (gfx1250) — compile-verified
//
#include <hip/hip_runtime.h>

// ---------------------------------------------------------------------------
// Gray-Scott residual (f_u, f_v) — memory-bound stencil.
// CDNA5 path: TDM tensor_load_to_lds stages 20x68 halo tiles into LDS.
// ---------------------------------------------------------------------------

typedef void __attribute__((address_space(3))) as3_void;
typedef const void __attribute__((address_space(1))) as1_cvoid;
typedef unsigned int u32x4 __attribute__((ext_vector_type(4)));
typedef int i32x8 __attribute__((ext_vector_type(8)));
typedef int i32x4 __attribute__((ext_vector_type(4)));

#if defined(__has_builtin)
#  if __has_builtin(__builtin_amdgcn_tensor_load_to_lds) && __has_builtin(__builtin_amdgcn_s_wait_tensorcnt)
#    define USE_TDM 1
#  elif __has_builtin(__builtin_amdgcn_global_load_async_to_lds_b32) && __has_builtin(__builtin_amdgcn_s_wait_asynccnt)
#    define USE_ASYNC 1
#  endif
#endif

namespace {
constexpr int Tn = 20, Bn = 8, Hn = 256, Wn = 256;
constexpr int OHn = 252, OWn = 252;
constexpr int TILE_H = 16, TILE_W = 64;
constexpr int LH = TILE_H + 4;   // 20
constexpr int LW = TILE_W + 4;   // 68
constexpr long long HW = (long long)Hn * Wn;          // 65536
constexpr long long TSTRIDE = (long long)Bn * 3 * HW; // elems between t and t+1, same (b,c)
}

__global__ __launch_bounds__(256)
void gs_residual_kernel(const float* __restrict__ src,
                        float* __restrict__ fu,
                        float* __restrict__ fv)
{
    __shared__ float su[LH * LW];
    __shared__ float sv[LH * LW];

    const int tid = threadIdx.x;
    const int tw  = blockIdx.x * TILE_W;   // output-tile origin (cols)
    const int th  = blockIdx.y * TILE_H;   // output-tile origin (rows)
    const int tb  = blockIdx.z;            // t*Bn + b
    const int t   = tb / Bn;

    const float* uplane = src + ((long long)tb * 3 + 1) * HW; // channel 1
    const float* vplane = uplane + HW;                        // channel 2

    // ---- Stage 20x68 halo tiles of u and v into LDS --------------------
#if defined(USE_TDM)
    if (tid == 0) {
        const unsigned lds_u = (unsigned)(size_t)(as3_void*)su;
        const unsigned lds_v = (unsigned)(size_t)(as3_void*)sv;
        const unsigned long long ga_u =
            (unsigned long long)(const void*)(uplane + (long long)th * Wn + tw);
        const unsigned long long ga_v = ga_u + (unsigned long long)HW * 4ull;
        const unsigned td0 = (unsigned)(Wn - tw);  // remaining cols (OOB -> 0)
        const unsigned td1 = (unsigned)(Hn - th);  // remaining rows (OOB -> 0)

        // D# group0: count=1 | lds_addr | global_addr[56:0] | type=2
        u32x4 g0u = { 1u, lds_u, (unsigned)ga_u,
                      (unsigned)((ga_u >> 32) & 0x01FFFFFFu) | (2u << 30) };
        u32x4 g0v = { 1u, lds_v, (unsigned)ga_v,
                      (unsigned)((ga_v >> 32) & 0x01FFFFFFu) | (2u << 30) };
        // D# group1: wg_mask=0, data_size=4B(code 2), dims/strides/tile dims
        i32x8 g1 = {
            (int)0x20000u,                                  // data_size=2 (4 bytes)
            (int)((td0 & 0xFFFFu) << 16),                   // abar=0 | tdim0 lo
            (int)(((td0 >> 16) & 0xFFFFu) | ((td1 & 0xFFFFu) << 16)),
            (int)(((td1 >> 16) & 0xFFFFu) | ((unsigned)LW << 16)), // tile_dim0=68
            (int)LH,                                        // tile_dim1=20, tile_dim2=0
            (int)Wn,                                        // tensor_dim0_stride=256
            0, 0                                            // dim1_stride unused (2D)
        };
        i32x4 z4 = { 0, 0, 0, 0 };
        i32x8 z8 = { 0, 0, 0, 0, 0, 0, 0, 0 };
        __builtin_amdgcn_tensor_load_to_lds(g0u, g1, z4, z4, z8, 0);
        __builtin_amdgcn_tensor_load_to_lds(g0v, g1, z4, z4, z8, 0);
        __builtin_amdgcn_s_wait_tensorcnt(0);
    }
    __syncthreads();
#else
    for (int p = tid; p < LH * LW; p += 256) {
        const int r  = p / LW;
        const int c  = p - r * LW;
        const int gr = (th + r < Hn - 1) ? th + r : Hn - 1;  // clamp; clamped vals unused
        const int gc = (tw + c < Wn - 1) ? tw + c : Wn - 1;
        const long long g = (long long)gr * Wn + gc;
#if defined(USE_ASYNC)
        __builtin_amdgcn_global_load_async_to_lds_b32(
            (as1_cvoid*)(uplane + g), (as3_void*)(su + p), 0, 0);
        __builtin_amdgcn_global_load_async_to_lds_b32(
            (as1_cvoid*)(vplane + g), (as3_void*)(sv + p), 0, 0);
#else
        su[p] = uplane[g];
        sv[p] = vplane[g];
#endif
    }
#if defined(USE_ASYNC)
    __builtin_amdgcn_s_wait_asynccnt(0);
#endif
    __syncthreads();
#endif

    // ---- Compute -------------------------------------------------------
    const float inv_dx2 = 1.0f / (0.15625f * 0.15625f);  // 1/DX^2 = 40.96
    for (int p = tid; p < TILE_H * TILE_W; p += 256) {
        const int r  = p >> 6;        // / TILE_W
        const int c  = p & 63;        // % TILE_W
        const int oh = th + r;
        const int ow = tw + c;
        if (oh >= OHn || ow >= OWn) continue;

        const int li = (r + 2) * LW + (c + 2);
        const float uc = su[li];
        const float vc = sv[li];

        float lap_u = (4.0f / 3.0f) * (su[li - 1] + su[li + 1] + su[li - LW] + su[li + LW])
                    - (1.0f / 12.0f) * (su[li - 2] + su[li + 2] + su[li - 2 * LW] + su[li + 2 * LW])
                    - 5.0f * uc;
        float lap_v = (4.0f / 3.0f) * (sv[li - 1] + sv[li + 1] + sv[li - LW] + sv[li + LW])
                    - (1.0f / 12.0f) * (sv[li - 2] + sv[li + 2] + sv[li - 2 * LW] + sv[li + 2 * LW])
                    - 5.0f * vc;
        lap_u *= inv_dx2;
        lap_v *= inv_dx2;

        // temporal derivative (2nd-order central; one-sided at t=0, t=T-1)
        const long long cidx = (long long)(oh + 2) * Wn + (ow + 2);
        const float* ub = uplane + cidx;
        const float* vb = vplane + cidx;
        float u_t, v_t;
        if (t == 0) {
            u_t = (-1.5f * uc + 2.0f * ub[TSTRIDE] - 0.5f * ub[2 * TSTRIDE]) * 20.0f;
            v_t = (-1.5f * vc + 2.0f * vb[TSTRIDE] - 0.5f * vb[2 * TSTRIDE]) * 20.0f;
        } else if (t == Tn - 1) {
            u_t = (0.5f * ub[-2 * TSTRIDE] - 2.0f * ub[-TSTRIDE] + 1.5f * uc) * 20.0f;
            v_t = (0.5f * vb[-2 * TSTRIDE] - 2.0f * vb[-TSTRIDE] + 1.5f * vc) * 20.0f;
        } else {
            u_t = (ub[TSTRIDE] - ub[-TSTRIDE]) * 10.0f;
            v_t = (vb[TSTRIDE] - vb[-TSTRIDE]) * 10.0f;
        }

        const float uvv = uc * vc * vc;
        const float Fu = 0.16f * lap_u - uvv + 0.06f * (1.0f - uc) - u_t;
        const float Fv = 0.08f * lap_v + uvv - (0.06f + 0.062f) * vc - v_t;

        const size_t oidx = (size_t)tb * (OHn * OWn) + (size_t)oh * OWn + ow;
        fu[oidx] = Fu;
        fv[oidx] = Fv;
    }
}

extern "C" void kernel_launch(void* const* d_in, const int* in_sizes, int n_in,
                              void* d_out, int out_size, void* d_ws, size_t ws_size,
                              hipStream_t stream)
{
    (void)in_sizes; (void)n_in; (void)d_ws; (void)ws_size; (void)out_size;
    const float* src = (const float*)d_in[0];
    float* fu = (float*)d_out;
    float* fv = fu + (size_t)Tn * Bn * OHn * OWn;   // second tuple element

    dim3 grid((OWn + TILE_W - 1) / TILE_W,   // 4
              (OHn + TILE_H - 1) / TILE_H,   // 16
              Tn * Bn);                      // 160
    gs_residual_kernel<<<grid, 256, 0, stream>>>(src, fu, fv);
}